// GNN_53996328845372
// MI455X (gfx1250) — compile-verified
//
#include <hip/hip_runtime.h>
#include <hip/hip_bf16.h>
#include <stdint.h>

#define N_NODES 50000
#define N_EDGES 600000
#define DIM     128
#define BN_EPS  1e-5f
#define SLOPE   0.01f
#define LDSS    132   // LDS row stride (floats): 4-bank rotation per row, 16B aligned

typedef __attribute__((ext_vector_type(2))) float v2f;
typedef __attribute__((ext_vector_type(8))) float v8f;

#if defined(__has_builtin)
#if __has_builtin(__builtin_amdgcn_wmma_f32_16x16x4_f32)
#define USE_WMMA_F32 1
#endif
#endif

// ---------------------------------------------------------------- zero
__global__ void gnn_zero_kernel(float* __restrict__ p, int n4) {
  int i = blockIdx.x * blockDim.x + threadIdx.x;
  int stride = gridDim.x * blockDim.x;
  float4 z = make_float4(0.f, 0.f, 0.f, 0.f);
  for (; i < n4; i += stride) ((float4*)p)[i] = z;
}

// ------------------------------------------------- edge gather + scatter-add
// One wave32 per edge: lane L moves cols [4L, 4L+3] of the 128-wide row.
// agg fits in L2 (25.6MB of 192MB) so the f32 atomics resolve in L2 atomic units.
__global__ __launch_bounds__(256)
void gnn_scatter_kernel(const float* __restrict__ x,
                        const int* __restrict__ ei,
                        float* __restrict__ agg) {
  const int lane = threadIdx.x & 31;
  int wid = (blockIdx.x * blockDim.x + threadIdx.x) >> 5;
  const int nwaves = (gridDim.x * blockDim.x) >> 5;
  for (int e = wid; e < N_EDGES; e += nwaves) {
    const int s = ei[e];            // src row
    const int d = ei[N_EDGES + e];  // dst row
    const float4 v = ((const float4*)(x + (size_t)s * DIM))[lane];
    float* dst = agg + (size_t)d * DIM + lane * 4;
    atomicAdd(dst + 0, v.x);
    atomicAdd(dst + 1, v.y);
    atomicAdd(dst + 2, v.z);
    atomicAdd(dst + 3, v.w);
  }
}

// ---------------------------------------------------------------- dual GEMM
// out[m][n] = bias[n] + sum_k A1[m][k]*W1[n][k] (+ sum_k A2[m][k]*W2[n][k])
// Block: 256 thr = 8 waves; block owns 16 rows, wave w owns cols [16w,16w+15].
// A tiles staged in LDS; B fragments (W row-major float2) read from global.
// Optionally accumulates per-column sum / sumsq for fused BatchNorm.
__global__ __launch_bounds__(256)
void gnn_gemm_dual_kernel(const float* __restrict__ A1, const float* __restrict__ W1,
                          const float* __restrict__ bias,
                          const float* __restrict__ A2, const float* __restrict__ W2,
                          float* __restrict__ out,
                          float* __restrict__ colsum, float* __restrict__ colsumsq,
                          int with_stats, int with_A2) {
  __shared__ float ldsA[2][16 * LDSS];
  const int t  = threadIdx.x;
  const int m0 = blockIdx.x * 16;

  // cooperative load of the 16x128 activation tile(s): 512 float4, 2 per thread
  for (int i = 0; i < 2; ++i) {
    const int f   = t + 256 * i;   // float4 index 0..511
    const int row = f >> 5;        // 32 float4 per row
    const int c4  = f & 31;
    float4 va = ((const float4*)(A1 + (size_t)(m0 + row) * DIM))[c4];
    *((float4*)&ldsA[0][row * LDSS + c4 * 4]) = va;
    if (with_A2) {
      float4 vb = ((const float4*)(A2 + (size_t)(m0 + row) * DIM))[c4];
      *((float4*)&ldsA[1][row * LDSS + c4 * 4]) = vb;
    }
  }
  __syncthreads();

  const int lane = t & 31;
  const int w    = t >> 5;
  const int n0   = w * 16;
  const int hl   = lane >> 4;   // half-wave: 0 -> K {0,1}/M rows 0..7 ; 1 -> K {2,3}/rows 8..15
  const int mr   = lane & 15;
  const int n    = n0 + mr;

  v8f acc;
  {
    const float bv = bias[n];
    for (int r = 0; r < 8; ++r) acc[r] = bv;
  }

  const float* w1row = W1 + (size_t)n * DIM;
#ifdef USE_WMMA_F32
  // V_WMMA_F32_16X16X4_F32 fragments:
  //   A: lane(hl,mr) holds A[mr][k+2*hl], A[mr][k+2*hl+1]
  //   B: lane(hl,mr) holds B[k+2*hl][n]=W[n][k+2*hl], B[k+2*hl+1][n]=W[n][k+2*hl+1]
  for (int k = 0; k < DIM; k += 4) {
    v2f a = *((const v2f*)&ldsA[0][mr * LDSS + k + 2 * hl]);
    v2f b;
    b.x = w1row[k + 2 * hl];
    b.y = w1row[k + 2 * hl + 1];
    acc = __builtin_amdgcn_wmma_f32_16x16x4_f32(false, a, false, b, (short)0, acc,
                                                false, false);
  }
  if (with_A2) {
    const float* w2row = W2 + (size_t)n * DIM;
    for (int k = 0; k < DIM; k += 4) {
      v2f a = *((const v2f*)&ldsA[1][mr * LDSS + k + 2 * hl]);
      v2f b;
      b.x = w2row[k + 2 * hl];
      b.y = w2row[k + 2 * hl + 1];
      acc = __builtin_amdgcn_wmma_f32_16x16x4_f32(false, a, false, b, (short)0, acc,
                                                  false, false);
    }
  }
#else
  // scalar fallback (also taken by the host pass of the compilation)
  for (int r = 0; r < 8; ++r) {
    const int m = 8 * hl + r;
    float s = acc[r];
    for (int k = 0; k < DIM; ++k) s += ldsA[0][m * LDSS + k] * w1row[k];
    if (with_A2) {
      const float* w2row = W2 + (size_t)n * DIM;
      for (int k = 0; k < DIM; ++k) s += ldsA[1][m * LDSS + k] * w2row[k];
    }
    acc[r] = s;
  }
#endif

  // store D tile + fused BN statistics.
  // C/D layout: lane(hl,mr), VGPR r -> out[m0 + 8*hl + r][n0 + mr]
  float s1 = 0.f, s2 = 0.f;
  for (int r = 0; r < 8; ++r) {
    const float v = acc[r];
    out[(size_t)(m0 + 8 * hl + r) * DIM + n] = v;
    s1 += v;
    s2 += v * v;
  }
  if (with_stats) {
    atomicAdd(&colsum[n], s1);
    atomicAdd(&colsumsq[n], s2);
  }
}

// -------------------------------------------- fold BN stats into scale/shift
__global__ void gnn_bn_prep_kernel(const float* __restrict__ colsum,
                                   const float* __restrict__ colsumsq,
                                   const float* __restrict__ g,
                                   const float* __restrict__ bt,
                                   float* __restrict__ scale,
                                   float* __restrict__ shift) {
  const int c = threadIdx.x;
  const float inv_n = 1.0f / (float)N_NODES;
  const float m  = colsum[c] * inv_n;
  const float v  = colsumsq[c] * inv_n - m * m;
  const float rs = rsqrtf(v + BN_EPS);
  const float sc = g[c] * rs;
  scale[c] = sc;
  shift[c] = bt[c] - m * sc;
}

// ------------------------------------------------- BN apply + LeakyReLU
__global__ __launch_bounds__(256)
void gnn_bn_lrelu_kernel(const float* __restrict__ h,
                         const float* __restrict__ scale,
                         const float* __restrict__ shift,
                         float* __restrict__ outp) {
  const int n4 = N_NODES * DIM / 4;
  int i = blockIdx.x * blockDim.x + threadIdx.x;
  const int stride = gridDim.x * blockDim.x;
  for (; i < n4; i += stride) {
    const int c = (i & 31) * 4;   // (4*i) % 128
    float4 v  = ((const float4*)h)[i];
    float4 sc = *((const float4*)&scale[c]);
    float4 sh = *((const float4*)&shift[c]);
    float4 o;
    o.x = v.x * sc.x + sh.x;
    o.y = v.y * sc.y + sh.y;
    o.z = v.z * sc.z + sh.z;
    o.w = v.w * sc.w + sh.w;
    o.x = (o.x >= 0.f) ? o.x : SLOPE * o.x;
    o.y = (o.y >= 0.f) ? o.y : SLOPE * o.y;
    o.z = (o.z >= 0.f) ? o.z : SLOPE * o.z;
    o.w = (o.w >= 0.f) ? o.w : SLOPE * o.w;
    ((float4*)outp)[i] = o;
  }
}

// ---------------------------------------------------------------- launcher
extern "C" void kernel_launch(void* const* d_in, const int* in_sizes, int n_in,
                              void* d_out, int out_size, void* d_ws, size_t ws_size,
                              hipStream_t stream) {
  (void)in_sizes; (void)n_in; (void)out_size; (void)ws_size;

  const float* x  = (const float*)d_in[0];
  const int*   ei = (const int*)d_in[1];
  const float* W_rel[3]  = {(const float*)d_in[2], (const float*)d_in[5], (const float*)d_in[8]};
  const float* b_rel[3]  = {(const float*)d_in[3], (const float*)d_in[6], (const float*)d_in[9]};
  const float* W_root[3] = {(const float*)d_in[4], (const float*)d_in[7], (const float*)d_in[10]};
  const float* g_[2]  = {(const float*)d_in[11], (const float*)d_in[13]};
  const float* bt_[2] = {(const float*)d_in[12], (const float*)d_in[14]};
  const float* W_lin = (const float*)d_in[15];
  const float* b_lin = (const float*)d_in[16];

  const size_t ND = (size_t)N_NODES * DIM;
  float* ws       = (float*)d_ws;
  float* agg      = ws;                 // [N, D]
  float* colsum   = agg + ND;           // [D]
  float* colsumsq = colsum + DIM;       // [D]
  float* scale    = colsumsq + DIM;     // [D]
  float* shift    = scale + DIM;        // [D]
  float* h_tmp    = shift + DIM;        // [N, D] pre-BN / layer-3 output
  float* hA       = h_tmp + ND;         // [N, D]
  float* hB       = hA + ND;            // [N, D]

  const dim3 blk(256);
  const int zgrid = 2048;
  const int sgrid = 2048;            // 16384 wave32 -> ~37 edges each
  const int ggrid = N_NODES / 16;    // 3125 row tiles
  const int egrid = 2048;
  const int zcount4 = (int)((ND + 2 * DIM) / 4);  // agg + colsum + colsumsq

  const float* hin = x;
  float* houts[2] = {hA, hB};

  for (int l = 0; l < 2; ++l) {
    gnn_zero_kernel<<<zgrid, blk, 0, stream>>>(agg, zcount4);
    gnn_scatter_kernel<<<sgrid, blk, 0, stream>>>(hin, ei, agg);
    gnn_gemm_dual_kernel<<<ggrid, blk, 0, stream>>>(agg, W_rel[l], b_rel[l],
                                                    hin, W_root[l], h_tmp,
                                                    colsum, colsumsq, 1, 1);
    gnn_bn_prep_kernel<<<1, DIM, 0, stream>>>(colsum, colsumsq, g_[l], bt_[l],
                                              scale, shift);
    gnn_bn_lrelu_kernel<<<egrid, blk, 0, stream>>>(h_tmp, scale, shift, houts[l]);
    hin = houts[l];
  }

  // layer 3 (no BN) then final linear
  gnn_zero_kernel<<<zgrid, blk, 0, stream>>>(agg, zcount4);
  gnn_scatter_kernel<<<sgrid, blk, 0, stream>>>(hin, ei, agg);
  gnn_gemm_dual_kernel<<<ggrid, blk, 0, stream>>>(agg, W_rel[2], b_rel[2],
                                                  hin, W_root[2], h_tmp,
                                                  colsum, colsumsq, 0, 1);
  gnn_gemm_dual_kernel<<<ggrid, blk, 0, stream>>>(h_tmp, W_lin, b_lin,
                                                  nullptr, nullptr, (float*)d_out,
                                                  colsum, colsumsq, 0, 0);
}